// WarpedSpaceKNN_36094905155941
// MI455X (gfx1250) — compile-verified
//
#include <hip/hip_runtime.h>
#include <hip/hip_bf16.h>

typedef __attribute__((ext_vector_type(2))) float v2f;
typedef __attribute__((ext_vector_type(8))) float v8f;

#define SEGS   2
#define NPTS   8192
#define KNN    65          // K+1 neighbours returned (self first)
#define FEAT   24          // 21 useful features padded to 24 (6 WMMA k-steps of 4)
#define CHUNK  512         // candidate columns buffered in LDS per iteration
#define DSTRIDE 516        // 512+4 pad: rows r and r+8 land in different banks
#define TPB    128         // 4 waves of 32
#define TOPP   72          // padded top-list stride

// ---------------------------------------------------------------------------
// Kernel 1: per-point featurization.
//   query features f_i  = [ vec(M^T M) (16), -2 M^T (M x) (4), ||Mx||^2, 0,0,0 ]
//   cand  features g_j  = [ vec(x x^T) (16),  x (4),            1,       0,0,0 ]
//   so dist(i,j) = f_i . g_j  exactly (fp32, same math as reference).
// ---------------------------------------------------------------------------
__global__ void knn_featurize(const float* __restrict__ xg,
                              const float* __restrict__ wg,
                              float* __restrict__ fq,
                              float* __restrict__ gc, int ntot) {
    int i = blockIdx.x * blockDim.x + threadIdx.x;
    if (i >= ntot) return;

    float x[4], M[4][4];
#pragma unroll
    for (int k = 0; k < 4; ++k) x[k] = xg[i * 4 + k];
#pragma unroll
    for (int k = 0; k < 4; ++k)
#pragma unroll
        for (int l = 0; l < 4; ++l) M[k][l] = wg[i * 16 + k * 4 + l];

    float a[4], aa = 0.f;
#pragma unroll
    for (int k = 0; k < 4; ++k) {
        float s = 0.f;
#pragma unroll
        for (int l = 0; l < 4; ++l) s += M[k][l] * x[l];
        a[k] = s; aa += s * s;
    }

    float* f = fq + (size_t)i * FEAT;
    float* g = gc + (size_t)i * FEAT;
#pragma unroll
    for (int s = 0; s < 4; ++s)
#pragma unroll
        for (int t = 0; t < 4; ++t) {
            float G = 0.f;
#pragma unroll
            for (int k = 0; k < 4; ++k) G += M[k][s] * M[k][t];
            f[s * 4 + t] = G;
            g[s * 4 + t] = x[s] * x[t];
        }
#pragma unroll
    for (int l = 0; l < 4; ++l) {
        float b = 0.f;
#pragma unroll
        for (int k = 0; k < 4; ++k) b += M[k][l] * a[k];
        f[16 + l] = -2.f * b;
        g[16 + l] = x[l];
    }
    f[20] = aa;  f[21] = 0.f; f[22] = 0.f; f[23] = 0.f;
    g[20] = 1.f; g[21] = 0.f; g[22] = 0.f; g[23] = 0.f;
}

// ---------------------------------------------------------------------------
// Kernel 2: fused WMMA distance tiles + streaming top-65 selection.
// One workgroup (4 waves) owns 16 query rows of one segment.
// ---------------------------------------------------------------------------
__global__ __launch_bounds__(TPB)
void knn_select(const float* __restrict__ fq,
                const float* __restrict__ gc,
                int* __restrict__ oidx,
                float* __restrict__ odist) {
    __shared__ float dch[16 * DSTRIDE];   // distance chunk  16 x 512 (padded)
    __shared__ float topd[16 * TOPP];     // sorted ascending distances
    __shared__ int   topi[16 * TOPP];     // matching global indices
    __shared__ int   cnt[16];

    const int tid   = threadIdx.x;
    const int lane  = tid & 31;
    const int wave  = tid >> 5;
    const int l15   = lane & 15;
    const int hi    = lane >> 4;          // 0 for lanes 0-15, 1 for 16-31
    const int qbase = blockIdx.x * 16;    // global query base
    const int seg   = qbase / NPTS;
    const int cseg0 = seg * NPTS;         // segment's first global point

    for (int e = tid; e < 16 * TOPP; e += TPB) { topd[e] = 3.0e38f; topi[e] = -1; }
    if (tid < 16) cnt[tid] = 0;
    __syncthreads();

    // --- A-matrix fragments (query features), resident for whole kernel ----
    // V_WMMA_F32_16X16X4_F32 A layout: lane m (0-15) holds K={0,1}, lane m+16
    // holds K={2,3} -> each lane loads 2 consecutive floats at k = 4s + 2*hi.
    const size_t arow = (size_t)(qbase + l15) * FEAT;
    v2f afrag[6];
#pragma unroll
    for (int s = 0; s < 6; ++s)
        afrag[s] = *(const v2f*)(fq + arow + s * 4 + 2 * hi);

    for (int cb = 0; cb < NPTS; cb += CHUNK) {
        // prefetch next candidate chunk's features into cache
        if (cb + CHUNK < NPTS) {
            const float* pf = gc + (size_t)(cseg0 + cb + CHUNK) * FEAT + tid * 96;
            __builtin_prefetch(pf, 0, 3);
        }

        // ---- compute 16 x CHUNK distance tile via WMMA chains -------------
#pragma unroll 1
        for (int p = 0; p < CHUNK / 64; ++p) {           // 8 passes, 4 waves x 16 cols
            const int ctile = cb + p * 64 + wave * 16;   // local candidate base
            const size_t brow = (size_t)(cseg0 + ctile + l15) * FEAT;
            v8f acc = {};
#pragma unroll
            for (int s = 0; s < 6; ++s) {
                v2f bfrag = *(const v2f*)(gc + brow + s * 4 + 2 * hi);
                acc = __builtin_amdgcn_wmma_f32_16x16x4_f32(
                        false, afrag[s], false, bfrag, (short)0, acc, false, false);
            }
            // D layout: VGPR r = row r (lanes 0-15) / row r+8 (lanes 16-31)
            const int col   = ctile + l15;               // local column
            const int rbase = hi * 8;
#pragma unroll
            for (int r = 0; r < 8; ++r) {
                const int row = r + rbase;
                float v = acc[r];
                if (qbase + row == cseg0 + col) v = -1.0f;   // self -> first
                dch[row * DSTRIDE + (col - cb)] = v;
            }
        }
        __syncthreads();

        // ---- merge chunk into per-row sorted top-65 ------------------------
        // wave w handles rows w, w+4, w+8, w+12.  Same-wave LDS ops are
        // in-order in hardware; the asm memory clobbers only stop the
        // compiler from hoisting/caching the threshold re-reads (keeps them
        // as cheap ds_load_b32 instead of volatile->flat SCOPE_SYS loads).
        for (int rr = 0; rr < 4; ++rr) {
            const int row = wave + rr * 4;
#pragma unroll 1
            for (int t = 0; t < CHUNK / 32; ++t) {
                asm volatile("" ::: "memory");           // re-read cnt/threshold
                const int   c = lane + t * 32;
                const float v = dch[row * DSTRIDE + c];
                const int   n = cnt[row];
                const float worst = topd[row * TOPP + KNN - 1];
                const bool pass = (n < KNN) || (v < worst);
                unsigned long long mask = __ballot(pass);
                while (mask) {
                    const int src = (int)__builtin_ctzll(mask);
                    mask &= mask - 1;
                    const float vv = __shfl(v, src, 32);
                    const int   cc = __shfl(c, src, 32);
                    if (lane == 0) {
                        asm volatile("" ::: "memory");
                        const int   n2 = cnt[row];
                        const float w2 = topd[row * TOPP + KNN - 1];
                        if (n2 < KNN || vv < w2) {
                            int pos = (n2 < KNN) ? n2 : (KNN - 1);
                            while (pos > 0 && topd[row * TOPP + pos - 1] > vv) {
                                topd[row * TOPP + pos] = topd[row * TOPP + pos - 1];
                                topi[row * TOPP + pos] = topi[row * TOPP + pos - 1];
                                --pos;
                            }
                            topd[row * TOPP + pos] = vv;
                            topi[row * TOPP + pos] = cseg0 + cb + cc; // global idx
                            if (n2 < KNN) cnt[row] = n2 + 1;
                        }
                        asm volatile("" ::: "memory");
                    }
                }
            }
        }
        __syncthreads();
    }

    // ---- emit results ------------------------------------------------------
    for (int e = tid; e < 16 * KNN; e += TPB) {
        const int row = e / KNN, k = e % KNN;
        const size_t o = (size_t)(qbase + row) * KNN + k;
        oidx[o]  = topi[row * TOPP + k];
        odist[o] = fmaxf(topd[row * TOPP + k], 0.0f);
    }
}

extern "C" void kernel_launch(void* const* d_in, const int* in_sizes, int n_in,
                              void* d_out, int out_size, void* d_ws, size_t ws_size,
                              hipStream_t stream) {
    const float* coords = (const float*)d_in[0];   // [S*N, 4]  fp32
    const float* warp   = (const float*)d_in[1];   // [S*N, 4, 4] fp32
    (void)in_sizes; (void)n_in; (void)out_size; (void)ws_size;

    const int ntot = SEGS * NPTS;
    float* fq = (float*)d_ws;                          // [ntot, FEAT]
    float* gc = fq + (size_t)ntot * FEAT;              // [ntot, FEAT]

    knn_featurize<<<(ntot + 127) / 128, 128, 0, stream>>>(coords, warp, fq, gc, ntot);

    int*   oidx  = (int*)d_out;                        // [ntot, 65] int32
    float* odist = (float*)d_out + (size_t)ntot * KNN; // [ntot, 65] fp32
    knn_select<<<ntot / 16, TPB, 0, stream>>>(fq, gc, oidx, odist);
}